// GATModel_43473658970189
// MI455X (gfx1250) — compile-verified
//
#include <hip/hip_runtime.h>
#include <math.h>

#define F_IN   128
#define HEADS  4
#define HID    64
#define C1     (HEADS * HID)   // 256
#define SLOPE  0.2f
#define EPS_SM 1e-16f

typedef float v2f __attribute__((ext_vector_type(2)));
typedef float v8f __attribute__((ext_vector_type(8)));

// ---------------------------------------------------------------------------
// fp32 GEMM: C[M,Nn] = A[M,K] @ B[K,Nn], row-major, via V_WMMA_F32_16X16X4_F32.
// One wave per 16x16 tile. M, Nn multiples of 16; K multiple of 4.
// A layout per lane (16x4 f32): lanes 0-15 hold K=0,1; lanes 16-31 hold K=2,3.
// B layout (4x16): VGPR0 = {K0 | K2}, VGPR1 = {K1 | K3} striped over lanes.
// C/D: 8 VGPRs, VGPR v -> M = v (lanes 0-15) / v+8 (lanes 16-31), N = lane%16.
// ---------------------------------------------------------------------------
__global__ void gemm_wmma_f32(const float* __restrict__ A,
                              const float* __restrict__ B,
                              float* __restrict__ C,
                              int M, int K, int Nn) {
  int wave = blockIdx.x * (blockDim.x >> 5) + (threadIdx.x >> 5);
  int lane = threadIdx.x & 31;
  int tiles_n = Nn >> 4;
  int tiles_m = M >> 4;
  int tile_m = wave / tiles_n;
  int tile_n = wave - tile_m * tiles_n;
  if (tile_m >= tiles_m) return;   // wave-uniform: EXEC stays all-1s for WMMA

  int m0 = tile_m << 4;
  int n0 = tile_n << 4;
  int mrow  = lane & 15;
  int khalf = (lane >> 4) << 1;    // 0 for lanes 0-15, 2 for lanes 16-31

  v8f acc = {};
  const float* arow = A + (size_t)(m0 + mrow) * K;
  for (int k = 0; k < K; k += 4) {
    v2f a, b;
    a.x = arow[k + khalf];
    a.y = arow[k + khalf + 1];
    b.x = B[(size_t)(k + khalf)     * Nn + n0 + mrow];
    b.y = B[(size_t)(k + khalf + 1) * Nn + n0 + mrow];
    acc = __builtin_amdgcn_wmma_f32_16x16x4_f32(
        /*neg_a=*/false, a, /*neg_b=*/false, b,
        /*c_mod=*/(short)0, acc, /*reuse_a=*/false, /*reuse_b=*/false);
  }

  int mo = m0 + ((lane >> 4) << 3);  // +8 for upper half-wave
#pragma unroll
  for (int v = 0; v < 8; ++v)
    C[(size_t)(mo + v) * Nn + n0 + (lane & 15)] = acc[v];
}

// ---------------------------------------------------------------------------
// Per-node attention logits: s[n,h] = <h[n,h,:], att_src[h,:]>, same for d.
// ---------------------------------------------------------------------------
__global__ void node_logits(const float* __restrict__ h,
                            const float* __restrict__ att_src,
                            const float* __restrict__ att_dst,
                            float* __restrict__ s, float* __restrict__ d,
                            int n, int heads, int hid) {
  int i = blockIdx.x * blockDim.x + threadIdx.x;
  if (i >= n * heads) return;
  int node = i / heads;
  int hh = i - node * heads;
  const float* hp = h + (size_t)node * heads * hid + hh * hid;
  const float* as = att_src + hh * hid;
  const float* ad = att_dst + hh * hid;
  float ss = 0.f, dd = 0.f;
#pragma unroll 4
  for (int c = 0; c < hid; ++c) {
    float v = hp[c];
    ss += v * as[c];
    dd += v * ad[c];
  }
  s[i] = ss;
  d[i] = dd;
}

// Monotonic float<->uint key (total order, handles negatives) for atomicMax.
__device__ __forceinline__ unsigned fkey(float f) {
  unsigned u = __float_as_uint(f);
  return u ^ ((unsigned)(((int)u) >> 31) | 0x80000000u);
}
__device__ __forceinline__ float funkey(unsigned k) {
  unsigned u = (k & 0x80000000u) ? (k ^ 0x80000000u) : ~k;
  return __uint_as_float(u);
}

__device__ __forceinline__ void edge_pair(const int* __restrict__ ei, int E_,
                                          int e, int* src, int* dst) {
  if (e < E_) { *src = ei[e]; *dst = ei[E_ + e]; }
  else        { *src = e - E_; *dst = e - E_; }   // self loop
}

__device__ __forceinline__ float lrelu(float x) {
  return x > 0.f ? x : SLOPE * x;
}

// Pass 1: segment max of leaky_relu(s[src]+d[dst]) per (dst, head).
__global__ void edge_max(const int* __restrict__ ei, int E_, int n,
                         const float* __restrict__ s, const float* __restrict__ d,
                         unsigned* __restrict__ mkey, int heads) {
  int i = blockIdx.x * blockDim.x + threadIdx.x;
  int tot = (E_ + n) * heads;
  if (i >= tot) return;
  int e = i / heads, hh = i - e * heads;
  int src, dst; edge_pair(ei, E_, e, &src, &dst);
  float logit = lrelu(s[src * heads + hh] + d[dst * heads + hh]);
  atomicMax(&mkey[dst * heads + hh], fkey(logit));
}

// Pass 2: e = exp(logit - max); accumulate denom; stash un-normalized e.
__global__ void edge_exp(const int* __restrict__ ei, int E_, int n,
                         const float* __restrict__ s, const float* __restrict__ d,
                         const unsigned* __restrict__ mkey,
                         float* __restrict__ den, float* __restrict__ alpha,
                         int heads) {
  int i = blockIdx.x * blockDim.x + threadIdx.x;
  int tot = (E_ + n) * heads;
  if (i >= tot) return;
  int e = i / heads, hh = i - e * heads;
  int src, dst; edge_pair(ei, E_, e, &src, &dst);
  float logit = lrelu(s[src * heads + hh] + d[dst * heads + hh]);
  float ex = __expf(logit - funkey(mkey[dst * heads + hh]));
  atomicAdd(&den[dst * heads + hh], ex);
  alpha[i] = ex;
}

// Pass 3: normalize alpha in place (alpha lives in the output buffer).
__global__ void alpha_norm(float* __restrict__ alpha,
                           const int* __restrict__ ei, int E_, int n,
                           const float* __restrict__ den, int heads) {
  int i = blockIdx.x * blockDim.x + threadIdx.x;
  int tot = (E_ + n) * heads;
  if (i >= tot) return;
  int e = i / heads, hh = i - e * heads;
  int src, dst; edge_pair(ei, E_, e, &src, &dst);
  alpha[i] = alpha[i] / (den[dst * heads + hh] + EPS_SM);
}

// Pass 4: agg[dst, h, c] += alpha[e, h] * h[src, h, c]; one thread per (e, h*c).
__global__ void edge_scatter(const int* __restrict__ ei, int E_, int n,
                             const float* __restrict__ alpha,
                             const float* __restrict__ h,
                             float* __restrict__ agg, int heads, int hid) {
  int tid = blockIdx.x * blockDim.x + threadIdx.x;
  int HC = heads * hid;
  int tot = (E_ + n) * HC;
  if (tid >= tot) return;
  int e = tid / HC;
  int j = tid - e * HC;
  int src, dst; edge_pair(ei, E_, e, &src, &dst);
  float a = alpha[e * heads + j / hid];
  atomicAdd(&agg[(size_t)dst * HC + j], a * h[(size_t)src * HC + j]);
}

// LayerNorm over C channels + ELU. One block (C threads) per node.
__global__ void ln_elu(const float* __restrict__ in, const float* __restrict__ bias,
                       const float* __restrict__ g, const float* __restrict__ b,
                       float* __restrict__ out, int C) {
  __shared__ float red[256];
  int node = blockIdx.x;
  int c = threadIdx.x;
  float v = in[(size_t)node * C + c] + bias[c];
  red[c] = v;
  __syncthreads();
  for (int st = C >> 1; st > 0; st >>= 1) {
    if (c < st) red[c] += red[c + st];
    __syncthreads();
  }
  float mu = red[0] * (1.0f / C);
  __syncthreads();
  float dv = v - mu;
  red[c] = dv * dv;
  __syncthreads();
  for (int st = C >> 1; st > 0; st >>= 1) {
    if (c < st) red[c] += red[c + st];
    __syncthreads();
  }
  float var = red[0] * (1.0f / C);
  float y = dv * rsqrtf(var + 1e-5f) * g[c] + b[c];
  out[(size_t)node * C + c] = y > 0.f ? y : expm1f(y);   // ELU (alpha=1)
}

// Prediction head: relu(h @ hW1 + hb1) @ hW2 + hb2. One thread per node.
__global__ void head_mlp(const float* __restrict__ h,
                         const float* __restrict__ hW1, const float* __restrict__ hb1,
                         const float* __restrict__ hW2, const float* __restrict__ hb2,
                         float* __restrict__ out, int n) {
  int i = blockIdx.x * blockDim.x + threadIdx.x;
  if (i >= n) return;
  const float* hp = h + (size_t)i * HID;
  float acc = hb2[0];
#pragma unroll 2
  for (int j = 0; j < HID / 2; ++j) {
    float z = hb1[j];
#pragma unroll 4
    for (int k = 0; k < HID; ++k) z += hp[k] * hW1[k * (HID / 2) + j];
    acc += (z > 0.f ? z : 0.f) * hW2[j];
  }
  out[i] = acc;
}

// ---------------------------------------------------------------------------
extern "C" void kernel_launch(void* const* d_in, const int* in_sizes, int n_in,
                              void* d_out, int out_size, void* d_ws, size_t ws_size,
                              hipStream_t stream) {
  const float* x    = (const float*)d_in[0];
  const int*   ei   = (const int*)  d_in[1];
  const float* W1   = (const float*)d_in[2];
  const float* as1  = (const float*)d_in[3];
  const float* ad1  = (const float*)d_in[4];
  const float* b1   = (const float*)d_in[5];
  const float* W2   = (const float*)d_in[6];
  const float* as2  = (const float*)d_in[7];
  const float* ad2  = (const float*)d_in[8];
  const float* b2   = (const float*)d_in[9];
  const float* ln1g = (const float*)d_in[10];
  const float* ln1b = (const float*)d_in[11];
  const float* ln2g = (const float*)d_in[12];
  const float* ln2b = (const float*)d_in[13];
  const float* hW1  = (const float*)d_in[14];
  const float* hb1  = (const float*)d_in[15];
  const float* hW2  = (const float*)d_in[16];
  const float* hb2  = (const float*)d_in[17];

  const int N  = in_sizes[0] / F_IN;   // 50000
  const int E  = in_sizes[1] / 2;      // 800000
  const int Et = E + N;                // with self loops

  // Workspace layout (floats)
  float* ws = (float*)d_ws;
  size_t o = 0;
  float* h1   = ws + o; o += (size_t)N * C1;    // [N,256] layer1 features (reused for LN1 out)
  float* agg1 = ws + o; o += (size_t)N * C1;    // [N,256] layer1 aggregation
  float* h2   = ws + o; o += (size_t)N * HID;   // [N,64]  layer2 features (reused for LN2 out)
  float* agg2 = ws + o; o += (size_t)N * HID;   // [N,64]  layer2 aggregation
  float*    s1   = ws + o; o += (size_t)N * HEADS;
  float*    dd1  = ws + o; o += (size_t)N * HEADS;
  unsigned* m1   = (unsigned*)(ws + o); o += (size_t)N * HEADS;
  float*    den1 = ws + o; o += (size_t)N * HEADS;
  float*    s2   = ws + o; o += (size_t)N;
  float*    dd2  = ws + o; o += (size_t)N;
  unsigned* m2   = (unsigned*)(ws + o); o += (size_t)N;
  float*    den2 = ws + o; o += (size_t)N;

  // Output layout: out [N] | alpha1 [Et*HEADS] | alpha2 [Et]
  float* outp   = (float*)d_out;
  float* alpha1 = outp + N;
  float* alpha2 = alpha1 + (size_t)Et * HEADS;

  // Deterministic init each call (graph-capture safe)
  hipMemsetAsync(agg1, 0, (size_t)N * C1 * sizeof(float), stream);
  hipMemsetAsync(agg2, 0, (size_t)N * HID * sizeof(float), stream);
  hipMemsetAsync(m1,   0, (size_t)N * HEADS * sizeof(unsigned), stream);
  hipMemsetAsync(den1, 0, (size_t)N * HEADS * sizeof(float), stream);
  hipMemsetAsync(m2,   0, (size_t)N * sizeof(unsigned), stream);
  hipMemsetAsync(den2, 0, (size_t)N * sizeof(float), stream);

  const int TB = 256;

  // ---- Layer 1 ----
  {
    int waves = (N >> 4) * (C1 >> 4);
    gemm_wmma_f32<<<(waves + 7) / 8, TB, 0, stream>>>(x, W1, h1, N, F_IN, C1);
  }
  node_logits<<<(N * HEADS + TB - 1) / TB, TB, 0, stream>>>(h1, as1, ad1, s1, dd1, N, HEADS, HID);
  {
    int tot = Et * HEADS;
    int blk = (tot + TB - 1) / TB;
    edge_max  <<<blk, TB, 0, stream>>>(ei, E, N, s1, dd1, m1, HEADS);
    edge_exp  <<<blk, TB, 0, stream>>>(ei, E, N, s1, dd1, m1, den1, alpha1, HEADS);
    alpha_norm<<<blk, TB, 0, stream>>>(alpha1, ei, E, N, den1, HEADS);
  }
  {
    long long tot = (long long)Et * C1;
    edge_scatter<<<(int)((tot + TB - 1) / TB), TB, 0, stream>>>(ei, E, N, alpha1, h1, agg1, HEADS, HID);
  }
  ln_elu<<<N, C1, 0, stream>>>(agg1, b1, ln1g, ln1b, h1, C1);   // h1 := LN1 output

  // ---- Layer 2 ----
  {
    int waves = (N >> 4) * (HID >> 4);
    gemm_wmma_f32<<<(waves + 7) / 8, TB, 0, stream>>>(h1, W2, h2, N, C1, HID);
  }
  node_logits<<<(N + TB - 1) / TB, TB, 0, stream>>>(h2, as2, ad2, s2, dd2, N, 1, HID);
  {
    int blk = (Et + TB - 1) / TB;
    edge_max  <<<blk, TB, 0, stream>>>(ei, E, N, s2, dd2, m2, 1);
    edge_exp  <<<blk, TB, 0, stream>>>(ei, E, N, s2, dd2, m2, den2, alpha2, 1);
    alpha_norm<<<blk, TB, 0, stream>>>(alpha2, ei, E, N, den2, 1);
  }
  {
    long long tot = (long long)Et * HID;
    edge_scatter<<<(int)((tot + TB - 1) / TB), TB, 0, stream>>>(ei, E, N, alpha2, h2, agg2, 1, HID);
  }
  ln_elu<<<N, HID, 0, stream>>>(agg2, b2, ln2g, ln2b, h2, HID); // h2 := LN2 output

  // ---- Head ----
  head_mlp<<<(N + TB - 1) / TB, TB, 0, stream>>>(h2, hW1, hb1, hW2, hb2, outp, N);
}